// LIFNeuron_61409442398455
// MI455X (gfx1250) — compile-verified
//
#include <hip/hip_runtime.h>
#include <stdint.h>

// LIF scan over T=32: spikes[t] = (mem >= 1), mem = 0.25*mem + x[t] - spike.
// HBM-bound (512 MB traffic, ~22us floor at 23.3 TB/s). Strategy:
//  - one thread per float4, serial loop over T (8 MB stride per step)
//  - 8-deep async global->LDS pipeline (gfx1250 global_load_async_to_lds_b128,
//    ASYNCcnt-tracked, per-wave private LDS slots so no barriers needed)
//  - th:TH_LOAD_NT on the read-once input stream (L2 = 192 MB < 512 MB stream)
//  - non-temporal b128 stores for the write-once output stream

typedef float f4 __attribute__((ext_vector_type(4)));

#define T_STEPS 32
#define BLOCK   256
#define NBUF    8   // pipeline depth-7 (NBUF-1 slabs in flight, ASYNCcnt is 6-bit)

__device__ __forceinline__ void async_load_b128(uint32_t lds_byte_addr, uint64_t gaddr) {
    // GV mode: VDST = LDS byte address, VADDR = 64-bit global address.
    // Non-temporal: read-once stream, do not retain in WGP$/L2.
    asm volatile("global_load_async_to_lds_b128 %0, %1, off th:TH_LOAD_NT"
                 :: "v"(lds_byte_addr), "v"(gaddr)
                 : "memory");
}

template <int N>
__device__ __forceinline__ void wait_asynccnt() {
    asm volatile("s_wait_asynccnt %0" :: "i"(N) : "memory");
}

// Drain the last K outstanding slabs with exact compile-time wait thresholds:
// Drain<K>: wait asynccnt <= K-1, compute, recurse.
template <int K>
struct Drain {
    template <typename F>
    static __device__ __forceinline__ void run(F& step, int t) {
        wait_asynccnt<K - 1>();
        step(t);
        Drain<K - 1>::run(step, t + 1);
    }
};
template <>
struct Drain<0> {
    template <typename F>
    static __device__ __forceinline__ void run(F&, int) {}
};

__global__ __launch_bounds__(BLOCK) void lif_scan_kernel(const float* __restrict__ x,
                                                         float* __restrict__ out,
                                                         unsigned spatial /* floats per timestep */) {
    __shared__ f4 sbuf[NBUF][BLOCK];

    const int tid = threadIdx.x;
    // byte offset of this thread's 16B element within a timestep slab
    const uint64_t base   = ((uint64_t)blockIdx.x * BLOCK + (unsigned)tid) * 16ull;
    const uint64_t stride = (uint64_t)spatial * 4ull;   // bytes per timestep (8 MB)

    uint64_t gload  = (uint64_t)(uintptr_t)x + base;
    char*    ostore = (char*)out + base;

    // Low 32 bits of the generic (flat) shared-aperture address == LDS byte offset.
    const uint32_t lds_base      = (uint32_t)(uintptr_t)&sbuf[0][tid];
    const uint32_t lds_bufstride = (uint32_t)(BLOCK * sizeof(f4)); // 4096 B

    // ---- prime pipeline: slabs 0..NBUF-2 in flight ----
#pragma unroll
    for (int i = 0; i < NBUF - 1; ++i) {
        async_load_b128(lds_base + (uint32_t)i * lds_bufstride, gload);
        gload += stride;
    }

    f4 mem = {0.f, 0.f, 0.f, 0.f};

    auto step = [&](int t) {
        f4 xv = sbuf[t & (NBUF - 1)][tid];   // ds_load_b128
        f4 sp;
#pragma unroll
        for (int k = 0; k < 4; ++k) {
            float m = __builtin_fmaf(0.25f, mem[k], xv[k]); // mem = beta*mem + x
            float s = (m >= 1.0f) ? 1.0f : 0.0f;            // Heaviside(m - 1)
            sp[k]  = s;
            mem[k] = m - s;                                  // soft reset
        }
        __builtin_nontemporal_store(sp, (f4*)ostore);        // write-once stream
        ostore += stride;
    };

    // ---- main loop: issue slab t+NBUF-1, wait for slab t, compute ----
    for (int t = 0; t <= T_STEPS - NBUF; ++t) {
        async_load_b128(lds_base + (uint32_t)((t + NBUF - 1) & (NBUF - 1)) * lds_bufstride, gload);
        gload += stride;
        wait_asynccnt<NBUF - 1>();  // oldest of the <=NBUF outstanding is done
        step(t);
    }
    // ---- epilogue: drain remaining NBUF-1 slabs with exact thresholds ----
    Drain<NBUF - 1>::run(step, T_STEPS - (NBUF - 1));
}

extern "C" void kernel_launch(void* const* d_in, const int* in_sizes, int n_in,
                              void* d_out, int out_size, void* d_ws, size_t ws_size,
                              hipStream_t stream) {
    (void)n_in; (void)out_size; (void)d_ws; (void)ws_size;
    const float* x   = (const float*)d_in[0];
    float*       out = (float*)d_out;

    const long long total   = (long long)in_sizes[0];   // 67,108,864
    const long long spatial = total / T_STEPS;          // 2,097,152 floats / timestep
    const long long nvec    = spatial / 4;              // 524,288 float4 lanes
    const unsigned  nblocks = (unsigned)((nvec + BLOCK - 1) / BLOCK); // 2048

    lif_scan_kernel<<<nblocks, BLOCK, 0, stream>>>(x, out, (unsigned)spatial);
}